// CTRNN_plus_output_11063835754824
// MI455X (gfx1250) — compile-verified
//
#include <hip/hip_runtime.h>

typedef __attribute__((ext_vector_type(2))) float v2f;
typedef __attribute__((ext_vector_type(8))) float v8f;

#define T_STEPS 2048
#define BATCH   512
#define HDIM    128
#define ALPHA   1e-4f          // DT/TAU = 0.01/100
#define LDS_STRIDE 132         // 128 + 4 pad: kills LDS bank conflicts on A loads

// ---------------------------------------------------------------------------
// Scan kernel: one workgroup per 16-batch tile, 8 waves; wave w owns h-columns
// [16w,16w+16). W_hh^T slab lives in registers; h state double-buffered in LDS.
// Per step: 32x V_WMMA_F32_16X16X4_F32 (4 independent accumulator chains),
// leaky-integrator update in VALU, publish h to LDS + rnn_output to global.
// ---------------------------------------------------------------------------
__global__ __launch_bounds__(256)
void ctrnn_scan_kernel(const float* __restrict__ x,
                       const float* __restrict__ W_ih,
                       const float* __restrict__ b_ih,
                       const float* __restrict__ W_hh,
                       const float* __restrict__ b_hh,
                       float* __restrict__ rnn_out)
{
    __shared__ float hbuf[2][16 * LDS_STRIDE];
    __shared__ float xbuf[2][16];

    const int tid  = threadIdx.x;
    const int wave = tid >> 5;          // 0..7
    const int lane = tid & 31;
    const int l16  = lane & 15;
    const int hi   = lane >> 4;         // 0: lanes 0-15, 1: lanes 16-31
    const int batch_base = blockIdx.x * 16;
    const int n_global   = wave * 16 + l16;   // this lane's h-column

    // Per-lane constants for the xi / bias terms (I == 1, O column fixed).
    const float bhh = b_hh[n_global];
    const float wih = W_ih[n_global];   // W_ih is (H,1)
    const float bih = b_ih[n_global];

    // Preload B = W_hh^T tile for this wave: all K=128, N = wave's 16 columns.
    // Assumed f32 4x16 B layout (mirror of documented 16x4 A layout):
    //   VGPR0: lanes0-15 -> K=k0,   lanes16-31 -> K=k0+2   (N = lane%16)
    //   VGPR1: lanes0-15 -> K=k0+1, lanes16-31 -> K=k0+3
    // B[k][n] = W_hh[n][k]  (row-major HxH), so k,k+1 are contiguous.
    v2f bmat[32];
#pragma unroll
    for (int kk = 0; kk < 32; ++kk) {
        const int k0 = kk * 4 + hi * 2;
        bmat[kk] = *(const v2f*)(W_hh + (size_t)n_global * HDIM + k0);
    }

    // h0 = 0
    for (int i = tid; i < 16 * LDS_STRIDE; i += 256) hbuf[0][i] = 0.0f;
    if (wave == 0 && lane < 16)
        xbuf[0][lane] = x[batch_base + lane];           // t = 0 slice
    v8f hreg = {};                                      // this wave's h tile

    __syncthreads();

    int cur = 0;
    for (int t = 0; t < T_STEPS; ++t) {
        const int nxt = cur ^ 1;

        // Prefetch next step's x slice into the other buffer.
        if (wave == 0 && lane < 16 && (t + 1) < T_STEPS)
            xbuf[nxt][lane] = x[(size_t)(t + 1) * BATCH + batch_base + lane];

        // C = h_prev @ W_hh^T for this wave's 16x16 tile, K = 128.
        // 4 independent accumulators -> dependent chain of 8 WMMAs, not 32.
        v8f c0 = {}, c1 = {}, c2 = {}, c3 = {};
#pragma unroll
        for (int kk = 0; kk < 32; kk += 4) {
            const int kb = kk * 4 + hi * 2;
            v2f a0 = *(const v2f*)&hbuf[cur][l16 * LDS_STRIDE + kb];
            v2f a1 = *(const v2f*)&hbuf[cur][l16 * LDS_STRIDE + kb + 4];
            v2f a2 = *(const v2f*)&hbuf[cur][l16 * LDS_STRIDE + kb + 8];
            v2f a3 = *(const v2f*)&hbuf[cur][l16 * LDS_STRIDE + kb + 12];
            c0 = __builtin_amdgcn_wmma_f32_16x16x4_f32(false, a0, false, bmat[kk + 0],
                                                       (short)0, c0, false, false);
            c1 = __builtin_amdgcn_wmma_f32_16x16x4_f32(false, a1, false, bmat[kk + 1],
                                                       (short)0, c1, false, false);
            c2 = __builtin_amdgcn_wmma_f32_16x16x4_f32(false, a2, false, bmat[kk + 2],
                                                       (short)0, c2, false, false);
            c3 = __builtin_amdgcn_wmma_f32_16x16x4_f32(false, a3, false, bmat[kk + 3],
                                                       (short)0, c3, false, false);
        }
        v8f c = (c0 + c1) + (c2 + c3);

        // Leaky-integrator update + publish (C/D layout: VGPR r -> M = r + 8*hi,
        // N = lane%16).
        float* hdst = &hbuf[nxt][0];
        float* rnn_base =
            rnn_out + ((size_t)t * BATCH + batch_base) * HDIM + n_global;
#pragma unroll
        for (int r = 0; r < 8; ++r) {
            const int m = r + hi * 8;
            const float xi  = xbuf[cur][m] * wih + bih;
            const float pre = c[r] + xi + bhh;
            const float hn  = pre > 0.0f ? pre : 0.0f;
            const float h   = hreg[r] * (1.0f - ALPHA) + hn * ALPHA;
            hreg[r] = h;
            hdst[m * LDS_STRIDE + n_global] = h;        // publish for next step
            rnn_base[(size_t)m * HDIM]      = h;        // rnn_output[t]
        }

        __syncthreads();   // orders hbuf/xbuf double-buffer ping-pong
        cur = nxt;
    }
}

// ---------------------------------------------------------------------------
// FC head: out[t,b] = relu(dot(rnn[t,b,:], W_fc) + b_fc). Bandwidth bound
// (~537 MB read @ 23.3 TB/s ~= 23 us). Wave per row, float4 loads, shfl tree.
// ---------------------------------------------------------------------------
__global__ __launch_bounds__(256)
void ctrnn_fc_kernel(const float* __restrict__ rnn,
                     const float* __restrict__ W_fc,
                     const float* __restrict__ b_fc,
                     float* __restrict__ out)
{
    const int lane = threadIdx.x & 31;
    const int wave = threadIdx.x >> 5;
    const size_t row = (size_t)blockIdx.x * 8 + wave;   // T*B rows, exact fit

    const float4 v = *(const float4*)(rnn + row * HDIM + lane * 4);
    const float4 w = *(const float4*)(W_fc + lane * 4);
    float p = v.x * w.x + v.y * w.y + v.z * w.z + v.w * w.w;
#pragma unroll
    for (int off = 16; off >= 1; off >>= 1)
        p += __shfl_xor(p, off, 32);
    if (lane == 0) {
        const float o = p + b_fc[0];
        out[row] = o > 0.0f ? o : 0.0f;
    }
}

extern "C" void kernel_launch(void* const* d_in, const int* in_sizes, int n_in,
                              void* d_out, int out_size, void* d_ws, size_t ws_size,
                              hipStream_t stream)
{
    const float* x    = (const float*)d_in[0];
    const float* W_ih = (const float*)d_in[1];
    const float* b_ih = (const float*)d_in[2];
    const float* W_hh = (const float*)d_in[3];
    const float* b_hh = (const float*)d_in[4];
    const float* W_fc = (const float*)d_in[5];
    const float* b_fc = (const float*)d_in[6];

    float* out = (float*)d_out;                        // (T,B,1) first
    float* rnn = out + (size_t)T_STEPS * BATCH;        // then (T,B,H)

    ctrnn_scan_kernel<<<BATCH / 16, 256, 0, stream>>>(x, W_ih, b_ih, W_hh,
                                                      b_hh, rnn);
    ctrnn_fc_kernel<<<(T_STEPS * BATCH) / 8, 256, 0, stream>>>(rnn, W_fc,
                                                               b_fc, out);
}